// HarmonicAngleTorchForce_75419625717904
// MI455X (gfx1250) — compile-verified
//
#include <hip/hip_runtime.h>

typedef __attribute__((ext_vector_type(2))) float v2f;
typedef __attribute__((ext_vector_type(8))) float v8f;

#define NB1 2048
#define BS  256

// Kernel 1: per-angle energy, grid-stride; wave reduction via V_WMMA_F32_16X16X4_F32;
// one deterministic partial per block into d_ws.
__global__ __launch_bounds__(BS) void angle_energy_kernel(
    const float* __restrict__ coords,
    const int*   __restrict__ angles,
    const float* __restrict__ th0,
    const float* __restrict__ kk,
    float*       __restrict__ block_out,
    int n)
{
    float acc = 0.0f;
    const int stride = gridDim.x * blockDim.x;
    for (int t = blockIdx.x * blockDim.x + threadIdx.x; t < n; t += stride) {
        const int ai = angles[3 * t + 0];
        const int aj = angles[3 * t + 1];
        const int ak = angles[3 * t + 2];
        const float* pi = coords + 3 * ai;
        const float* pj = coords + 3 * aj;
        const float* pk = coords + 3 * ak;

        const float v1x = pi[0] - pj[0];
        const float v1y = pi[1] - pj[1];
        const float v1z = pi[2] - pj[2];
        const float v2x = pk[0] - pj[0];
        const float v2y = pk[1] - pj[1];
        const float v2z = pk[2] - pj[2];

        const float dot = v1x * v2x + v1y * v2y + v1z * v2z;
        const float cx  = v1y * v2z - v1z * v2y;
        const float cy  = v1z * v2x - v1x * v2z;
        const float cz  = v1x * v2y - v1y * v2x;
        const float cn  = sqrtf(cx * cx + cy * cy + cz * cz);

        const float theta = atan2f(cn, dot);
        const float dt    = theta - th0[t];
        acc = fmaf(0.5f * kk[t], dt * dt, acc);
    }

    // ---- Wave-level reduction via WMMA (f32, exact) ----
    // A (16x4 f32): each lane contributes its partial in A-VGPR0, 0 in A-VGPR1.
    // B = all ones (layout-invariant).  D[m][n] = rowsum(A)[m] for all n.
    v2f a; a.x = acc;  a.y = 0.0f;
    v2f b; b.x = 1.0f; b.y = 1.0f;
    v8f c = {};
    c = __builtin_amdgcn_wmma_f32_16x16x4_f32(
            /*neg_a=*/false, a, /*neg_b=*/false, b,
            /*c_mod=*/(short)0, c, /*reuse_a=*/false, /*reuse_b=*/false);

    // lanes 0-15 hold rowsums of M=0..7 in c[0..7]; lanes 16-31 hold M=8..15.
    float s = c[0] + c[1] + c[2] + c[3] + c[4] + c[5] + c[6] + c[7];
    float wave_total = s + __shfl_xor(s, 16, 32);   // identical in all 32 lanes

    __shared__ float wsum[BS / 32];
    const int lane = threadIdx.x & 31;
    const int wid  = threadIdx.x >> 5;
    if (lane == 0) wsum[wid] = wave_total;
    __syncthreads();

    if (threadIdx.x == 0) {
        float tsum = 0.0f;
#pragma unroll
        for (int i = 0; i < BS / 32; ++i) tsum += wsum[i];
        block_out[blockIdx.x] = tsum;
    }
}

// Kernel 2: single-block fixed-order tree reduction -> deterministic scalar.
__global__ __launch_bounds__(BS) void finalize_kernel(
    const float* __restrict__ partials, int n, float* __restrict__ out)
{
    __shared__ float sm[BS];
    float acc = 0.0f;
    for (int i = threadIdx.x; i < n; i += BS) acc += partials[i];
    sm[threadIdx.x] = acc;
    __syncthreads();
    for (int s = BS / 2; s > 0; s >>= 1) {
        if (threadIdx.x < s) sm[threadIdx.x] += sm[threadIdx.x + s];
        __syncthreads();
    }
    if (threadIdx.x == 0) out[0] = sm[0];
}

extern "C" void kernel_launch(void* const* d_in, const int* in_sizes, int n_in,
                              void* d_out, int out_size, void* d_ws, size_t ws_size,
                              hipStream_t stream) {
    const float* coords = (const float*)d_in[0];
    // d_in[1] = box: present in the signature but unused by the reference math.
    const int*   angles = (const int*)d_in[2];
    const float* th0    = (const float*)d_in[3];
    const float* kk     = (const float*)d_in[4];
    const int n = in_sizes[3];   // N_ANGLES

    int nb = NB1;
    if (ws_size < (size_t)nb * sizeof(float)) {
        nb = (int)(ws_size / sizeof(float));
        if (nb < 1) nb = 1;
    }
    float* partials = (float*)d_ws;

    angle_energy_kernel<<<nb, BS, 0, stream>>>(coords, angles, th0, kk, partials, n);
    finalize_kernel<<<1, BS, 0, stream>>>(partials, nb, (float*)d_out);
}